// SpatialProcessor_3539053052702
// MI455X (gfx1250) — compile-verified
//
#include <hip/hip_runtime.h>

typedef float v2f __attribute__((ext_vector_type(2)));
typedef float v8f __attribute__((ext_vector_type(8)));

#define N_NODES 4096
#define HID 128
#define OUT_DIM 256
#define HEADS 4
#define NEG_SLOPE 0.2f
#define LN_EPS 1e-5f

// ---------------------------------------------------------------------------
// GEMM: C[M,N] = A[M,K] @ B[K,N], all f32, via V_WMMA_F32_16X16X4_F32.
// Block = 8 waves; each wave computes one 16x16 tile; block covers 16x128.
// A-frag: lane l holds rows (l&15), K = 2*(l>>4)+{0,1}  -> contiguous float2.
// B-frag: lane l holds col (l&15), K = 2*(l>>4)+{0,1}   -> two strided loads.
// ---------------------------------------------------------------------------
__global__ void gemm_wmma_f32(const float* __restrict__ A,
                              const float* __restrict__ B,
                              float* __restrict__ C,
                              int M, int K, int N) {
  const int lane = threadIdx.x & 31;
  const int wave = threadIdx.x >> 5;
  const int lm   = lane & 15;
  const int half = lane >> 4;
  const int n0 = blockIdx.x * 128 + wave * 16;
  const int m0 = blockIdx.y * 16;

  v8f acc = {};
  const float* ap = A + (size_t)(m0 + lm) * K + 2 * half;
  const float* bp = B + (size_t)(2 * half) * N + n0 + lm;
  for (int k = 0; k < K; k += 4) {
    float2 av = *(const float2*)ap;
    v2f a; a.x = av.x; a.y = av.y;
    v2f b; b.x = bp[0]; b.y = bp[N];
    acc = __builtin_amdgcn_wmma_f32_16x16x4_f32(false, a, false, b,
                                                (short)0, acc, false, false);
    ap += 4;
    bp += (size_t)4 * N;
  }
#pragma unroll
  for (int v = 0; v < 8; ++v)
    C[(size_t)(m0 + v + 8 * half) * N + n0 + lm] = acc[v];
}

// ---------------------------------------------------------------------------
// Per (node, head): e_src = <z_row, a_src[h]>, e_dst = <z_row, a_dst[h]>,
// plus E1=exp(e_src), E2=exp(0.2*e_src). Layout [h][n] for contiguous access.
// grid = N, block = 128 (wave w handles head w).
// ---------------------------------------------------------------------------
__global__ void prep_logits(const float* __restrict__ z,
                            const float* __restrict__ a_src,
                            const float* __restrict__ a_dst,
                            float* __restrict__ es, float* __restrict__ ed,
                            float* __restrict__ E1, float* __restrict__ E2,
                            int D) {
  const int n = blockIdx.x;
  const int h = threadIdx.x >> 5;
  const int lane = threadIdx.x & 31;
  const float* zr = z + (size_t)n * (HEADS * D) + h * D;
  const float* as = a_src + h * D;
  const float* ad = a_dst + h * D;
  float ssrc = 0.f, sdst = 0.f;
  for (int d = lane; d < D; d += 32) {
    float zv = zr[d];
    ssrc += zv * as[d];
    sdst += zv * ad[d];
  }
  for (int off = 16; off; off >>= 1) {
    ssrc += __shfl_xor(ssrc, off);
    sdst += __shfl_xor(sdst, off);
  }
  if (lane == 0) {
    int idx = h * N_NODES + n;
    es[idx] = ssrc;
    ed[idx] = sdst;
    E1[idx] = expf(ssrc);
    E2[idx] = expf(NEG_SLOPE * ssrc);
  }
}

// Per-head max of e_src (softmax stabilizer; lrelu is monotone so
// max_j lrelu(c+e_j) = lrelu(c+emax)). grid = HEADS, block = 256.
__global__ void head_max(const float* __restrict__ es, float* __restrict__ emax) {
  __shared__ float red[8];
  const int h = blockIdx.x;
  float m = -1e30f;
  for (int i = threadIdx.x; i < N_NODES; i += 256)
    m = fmaxf(m, es[h * N_NODES + i]);
  for (int off = 16; off; off >>= 1) m = fmaxf(m, __shfl_xor(m, off));
  if ((threadIdx.x & 31) == 0) red[threadIdx.x >> 5] = m;
  __syncthreads();
  if (threadIdx.x < 8) {
    m = red[threadIdx.x];
    for (int off = 4; off; off >>= 1) m = fmaxf(m, __shfl_xor(m, off));
    if (threadIdx.x == 0) emax[h] = m;
  }
}

// Per dst: M = lrelu(c+emax); P = exp(c-M); Q = exp(0.2c-M); T = exp(-c)
// so that w_ij = (E1_j >= T_i) ? P_i*E1_j : Q_i*E2_j  (no exp in inner loop).
__global__ void prep_dst(const float* __restrict__ ed,
                         const float* __restrict__ emax,
                         float* __restrict__ P, float* __restrict__ Q,
                         float* __restrict__ T) {
  const int idx = blockIdx.x * 256 + threadIdx.x;  // 0..H*N-1
  const int h = idx / N_NODES;
  float c = ed[idx];
  float m = c + emax[h];
  m = m >= 0.f ? m : NEG_SLOPE * m;
  P[idx] = expf(c - m);
  Q[idx] = expf(NEG_SLOPE * c - m);
  T[idx] = expf(-c);
}

// ---------------------------------------------------------------------------
// Fused edge-softmax aggregation:
//   out[i, h*D+d] = (sum_j w_ij z[j,h,d]) / sum_j w_ij
// Block = 8 waves; wave owns one 16-dst tile, all D/16 d-tiles (f32 WMMA accs).
// Source z rows are staged 32-at-a-time into LDS with DOUBLE-BUFFERED
// GLOBAL_LOAD_ASYNC_TO_LDS_B128 (per-lane gather to padded LDS rows), so the
// next tile's HBM/L2->LDS copy overlaps this tile's 8*DT WMMAs.
// LDS row pad = 8 floats => the two K-half reads land on disjoint banks.
// ---------------------------------------------------------------------------
template <int D>
__global__ void attn_agg(const float* __restrict__ z,
                         const float* __restrict__ E1, const float* __restrict__ E2,
                         const float* __restrict__ P, const float* __restrict__ Q,
                         const float* __restrict__ T,
                         float* __restrict__ out) {
  constexpr int DT = D / 16;
  constexpr int LDSS = D + 8;
  constexpr int QPT = (32 * (D / 4)) / 256;  // async b128 ops per lane per tile
  __shared__ float sz[2][32 * LDSS];
  const int ldz = HEADS * D;
  const int h = blockIdx.y;
  const int wave = threadIdx.x >> 5;
  const int lane = threadIdx.x & 31;
  const int lm = lane & 15;
  const int half = lane >> 4;
  const int i0 = (blockIdx.x * 8 + wave) * 16;
  const int row = i0 + lm;

  const float* E1h = E1 + h * N_NODES;
  const float* E2h = E2 + h * N_NODES;
  const float p = P[h * N_NODES + row];
  const float q = Q[h * N_NODES + row];
  const float t = T[h * N_NODES + row];

  v8f acc[DT] = {};
  float dsum = 0.f;
  const float* zh = z + h * D;

  // issue the async gather of 32 src rows (tile at j0) into buffer `buf`
  auto issue_tile = [&](int j0, int buf) {
#pragma unroll
    for (int qi = 0; qi < QPT; ++qi) {
      int idx = (int)threadIdx.x + qi * 256;
      int jj = idx / (D / 4);
      int dq = idx % (D / 4);
      const float* gp = zh + (size_t)(j0 + jj) * ldz + dq * 4;
      unsigned lo = (unsigned)(size_t)(const void*)&sz[buf][jj * LDSS + dq * 4];
      asm volatile("global_load_async_to_lds_b128 %0, %1, off"
                   :: "v"(lo), "v"(gp) : "memory");
    }
  };

  constexpr int NT = N_NODES / 32;
  issue_tile(0, 0);
  for (int it = 0; it < NT; ++it) {
    const int buf = it & 1;
    if (it + 1 < NT) {
      issue_tile((it + 1) * 32, buf ^ 1);
      asm volatile("s_wait_asynccnt %0" :: "n"(QPT) : "memory");
    } else {
      asm volatile("s_wait_asynccnt 0" ::: "memory");
    }
    __syncthreads();

    const int j0 = it * 32;
    const float* sb = &sz[buf][0];
#pragma unroll
    for (int kc = 0; kc < 8; ++kc) {
      const int jl = j0 + kc * 4 + 2 * half;
      float2 e1 = *(const float2*)(E1h + jl);
      float2 e2 = *(const float2*)(E2h + jl);
      float w0 = (e1.x >= t) ? p * e1.x : q * e2.x;
      float w1 = (e1.y >= t) ? p * e1.y : q * e2.y;
      dsum += w0 + w1;
      v2f a; a.x = w0; a.y = w1;
      const float* b0 = sb + (kc * 4 + 2 * half) * LDSS + lm;
      const float* b1 = b0 + LDSS;
#pragma unroll
      for (int dt = 0; dt < DT; ++dt) {
        v2f b; b.x = b0[dt * 16]; b.y = b1[dt * 16];
        acc[dt] = __builtin_amdgcn_wmma_f32_16x16x4_f32(false, a, false, b,
                                                        (short)0, acc[dt],
                                                        false, false);
      }
    }
    __syncthreads();  // protect buffer `buf` before it is re-filled (it+2)
  }

  // softmax denominator: combine K-halves; lane l now has row (l&15) total
  dsum += __shfl_xor(dsum, 16);
  float inv = 1.0f / dsum;
#pragma unroll
  for (int v = 0; v < 8; ++v) {
    float rinv = __shfl(inv, v + 8 * half);  // row v + 8*half
    float* op = out + (size_t)(i0 + v + 8 * half) * ldz + h * D + lm;
#pragma unroll
    for (int dt = 0; dt < DT; ++dt) op[dt * 16] = acc[dt][v] * rinv;
  }
}

// ---------------------------------------------------------------------------
// In-place LayerNorm with pre-bias: x = LN(x + bias)*g + be, optional ReLU.
// grid = N, block = 256 (L = 512 or 1024).
// ---------------------------------------------------------------------------
__global__ void layernorm_bias(float* __restrict__ x, const float* __restrict__ bias,
                               const float* __restrict__ g, const float* __restrict__ be,
                               int L, int do_relu) {
  __shared__ float red0[8];
  __shared__ float red1[8];
  const int n = blockIdx.x;
  float* xr = x + (size_t)n * L;
  const int cnt = L / 256;
  float v[4];
  float s = 0.f;
  for (int i = 0; i < cnt; ++i) {
    int c = threadIdx.x + i * 256;
    v[i] = xr[c] + bias[c];
    s += v[i];
  }
  for (int off = 16; off; off >>= 1) s += __shfl_xor(s, off);
  if ((threadIdx.x & 31) == 0) red0[threadIdx.x >> 5] = s;
  __syncthreads();
  float tot = 0.f;
#pragma unroll
  for (int w = 0; w < 8; ++w) tot += red0[w];
  const float mean = tot / (float)L;
  float ss = 0.f;
  for (int i = 0; i < cnt; ++i) { float d = v[i] - mean; ss += d * d; }
  for (int off = 16; off; off >>= 1) ss += __shfl_xor(ss, off);
  if ((threadIdx.x & 31) == 0) red1[threadIdx.x >> 5] = ss;
  __syncthreads();
  float var = 0.f;
#pragma unroll
  for (int w = 0; w < 8; ++w) var += red1[w];
  var /= (float)L;
  const float rstd = rsqrtf(var + LN_EPS);
  for (int i = 0; i < cnt; ++i) {
    int c = threadIdx.x + i * 256;
    float y = (v[i] - mean) * rstd * g[c] + be[c];
    if (do_relu) y = fmaxf(y, 0.f);
    xr[c] = y;
  }
}

// ---------------------------------------------------------------------------
extern "C" void kernel_launch(void* const* d_in, const int* in_sizes, int n_in,
                              void* d_out, int out_size, void* d_ws, size_t ws_size,
                              hipStream_t stream) {
  (void)in_sizes; (void)n_in; (void)out_size; (void)ws_size;
  const float* x      = (const float*)d_in[0];
  /* d_in[1] = embed: structural only (complete graph) — unused */
  const float* W1     = (const float*)d_in[2];
  const float* a_src1 = (const float*)d_in[3];
  const float* a_dst1 = (const float*)d_in[4];
  const float* b1     = (const float*)d_in[5];
  const float* g1     = (const float*)d_in[6];
  const float* be1    = (const float*)d_in[7];
  const float* W2     = (const float*)d_in[8];
  const float* a_src2 = (const float*)d_in[9];
  const float* a_dst2 = (const float*)d_in[10];
  const float* b2     = (const float*)d_in[11];
  const float* g2     = (const float*)d_in[12];
  const float* be2    = (const float*)d_in[13];

  char* ws = (char*)d_ws;
  float* z1 = (float*)(ws);                        // 4096*512  f32 (8 MB)
  float* hb = (float*)(ws + 8ull * 1024 * 1024);   // 4096*512  f32 (8 MB)
  float* z2 = (float*)(ws + 16ull * 1024 * 1024);  // 4096*1024 f32 (16 MB)
  char* sm  = ws + 32ull * 1024 * 1024;
  const size_t SA = (size_t)N_NODES * HEADS * sizeof(float);  // 64 KB each
  float* es1 = (float*)(sm +  0 * SA);
  float* ed1 = (float*)(sm +  1 * SA);
  float* E1a = (float*)(sm +  2 * SA);
  float* E2a = (float*)(sm +  3 * SA);
  float* P1  = (float*)(sm +  4 * SA);
  float* Q1  = (float*)(sm +  5 * SA);
  float* T1  = (float*)(sm +  6 * SA);
  float* es2 = (float*)(sm +  7 * SA);
  float* ed2 = (float*)(sm +  8 * SA);
  float* E1b = (float*)(sm +  9 * SA);
  float* E2b = (float*)(sm + 10 * SA);
  float* P2  = (float*)(sm + 11 * SA);
  float* Q2  = (float*)(sm + 12 * SA);
  float* T2  = (float*)(sm + 13 * SA);
  float* emax = (float*)(sm + 14 * SA);            // 8 floats
  float* outp = (float*)d_out;

  // ---- Layer 1 ----
  gemm_wmma_f32<<<dim3(512 / 128, N_NODES / 16), 256, 0, stream>>>(
      x, W1, z1, N_NODES, 128, 512);
  prep_logits<<<N_NODES, 128, 0, stream>>>(z1, a_src1, a_dst1, es1, ed1, E1a, E2a, HID);
  head_max<<<HEADS, 256, 0, stream>>>(es1, emax);
  prep_dst<<<(N_NODES * HEADS) / 256, 256, 0, stream>>>(ed1, emax, P1, Q1, T1);
  attn_agg<HID><<<dim3(N_NODES / 128, HEADS), 256, 0, stream>>>(
      z1, E1a, E2a, P1, Q1, T1, hb);
  layernorm_bias<<<N_NODES, 256, 0, stream>>>(hb, b1, g1, be1, 512, 1);

  // ---- Layer 2 ----
  gemm_wmma_f32<<<dim3(1024 / 128, N_NODES / 16), 256, 0, stream>>>(
      hb, W2, z2, N_NODES, 512, 1024);
  prep_logits<<<N_NODES, 128, 0, stream>>>(z2, a_src2, a_dst2, es2, ed2, E1b, E2b, OUT_DIM);
  head_max<<<HEADS, 256, 0, stream>>>(es2, emax + 4);
  prep_dst<<<(N_NODES * HEADS) / 256, 256, 0, stream>>>(ed2, emax + 4, P2, Q2, T2);
  attn_agg<OUT_DIM><<<dim3(N_NODES / 128, HEADS), 256, 0, stream>>>(
      z2, E1b, E2b, P2, Q2, T2, outp);
  layernorm_bias<<<N_NODES, 256, 0, stream>>>(outp, b2, g2, be2, 1024, 0);
}